// TomoEmbedding_69329362092736
// MI455X (gfx1250) — compile-verified
//
#include <hip/hip_runtime.h>
#include <hip/hip_bf16.h>

// Sizes from the reference
#define C_    32
#define LM1_  2047
#define D_    512
#define H_    128
#define NBINS 32
#define ROWS_ 2048   // 1 + LM1_

typedef __attribute__((ext_vector_type(2))) float v2f;
typedef __attribute__((ext_vector_type(4))) float f4;
typedef __attribute__((ext_vector_type(8))) float v8f;

// ---------------------------------------------------------------------------
// Kernel 1: enc[32][512] = relu(bin_value * W1 + b1) @ W2 + b2  via WMMA f32.
// One wave (32 threads) computes one 16x16 tile of enc; K=128 in steps of 4.
// A-fragment 16x4 f32 layout: lanes 0-15 hold K=k0,k0+1; lanes 16-31 K=k0+2,k0+3.
// B-fragment 4x16 mirrors the K assignment; N = n0 + (lane&15).
// ---------------------------------------------------------------------------
__global__ __launch_bounds__(32) void enc_wmma_kernel(
    const float* __restrict__ W1, const float* __restrict__ b1,
    const float* __restrict__ W2, const float* __restrict__ b2,
    float* __restrict__ enc)
{
    const int lane = threadIdx.x & 31;
    const int half = lane >> 4;          // 0: K=k0..k0+1, 1: K=k0+2..k0+3
    const int l16  = lane & 15;
    const int m0   = (blockIdx.x & 1) * 16;    // M tiles: 32/16 = 2
    const int n0   = (blockIdx.x >> 1) * 16;   // N tiles: 512/16 = 32
    const int m    = m0 + l16;                 // bin value 0..31
    const int n    = n0 + l16;
    const float mf = (float)m;

    v8f acc = {0.f, 0.f, 0.f, 0.f, 0.f, 0.f, 0.f, 0.f};
    for (int k0 = 0; k0 < H_; k0 += 4) {
        const int kA = k0 + 2 * half;
        v2f a, bf;
        // h[m][k] = relu(m * W1[k] + b1[k]) computed on the fly
        a.x = fmaxf(fmaf(mf, W1[kA],     b1[kA]),     0.f);
        a.y = fmaxf(fmaf(mf, W1[kA + 1], b1[kA + 1]), 0.f);
        bf.x = W2[kA * D_ + n];
        bf.y = W2[(kA + 1) * D_ + n];
        acc = __builtin_amdgcn_wmma_f32_16x16x4_f32(
            /*neg_a=*/false, a, /*neg_b=*/false, bf,
            /*c_mod=*/(short)0, acc, /*reuse_a=*/false, /*reuse_b=*/false);
    }
    const float bias = b2[n];
    // C/D layout: VGPR v -> M = m0 + v + 8*half (lanes split), N = n0 + (lane&15)
#pragma unroll
    for (int v = 0; v < 8; ++v) {
        const int row = m0 + v + 8 * half;
        enc[row * D_ + n] = acc[v] + bias;
    }
}

// ---------------------------------------------------------------------------
// Kernel 2: logits = enc @ bins^T (31 bins), softmax, table[b] = probs @ bins,
// table[0] = bin_embeddings[0]. One block, all data tiny / cache resident.
// ---------------------------------------------------------------------------
__global__ __launch_bounds__(1024) void softmax_table_kernel(
    const float* __restrict__ enc, const float* __restrict__ bin_emb,
    float* __restrict__ table)
{
    __shared__ float probs_s[NBINS][NBINS];  // [b][n], n<31 used (padded stride)
    const int t = threadIdx.x;

    // Phase 1: 32*31 dot products of length 512
    if (t < NBINS * 31) {
        const int b = t / 31, n = t % 31;
        const f4* er = (const f4*)(enc + (size_t)b * D_);
        const f4* br = (const f4*)(bin_emb + (size_t)(n + 1) * D_);
        float s = 0.f;
        for (int i = 0; i < D_ / 4; ++i) {
            const f4 e = er[i], bb = br[i];
            s = fmaf(e.x, bb.x, s); s = fmaf(e.y, bb.y, s);
            s = fmaf(e.z, bb.z, s); s = fmaf(e.w, bb.w, s);
        }
        probs_s[b][n] = s;
    }
    __syncthreads();

    // Phase 2: row softmax over 31 entries
    if (t < NBINS) {
        float mx = -3.402823466e38f;
        for (int n = 0; n < 31; ++n) mx = fmaxf(mx, probs_s[t][n]);
        float ex[31];
        float sum = 0.f;
        for (int n = 0; n < 31; ++n) { ex[n] = expf(probs_s[t][n] - mx); sum += ex[n]; }
        const float inv = 1.f / sum;
        for (int n = 0; n < 31; ++n) probs_s[t][n] = ex[n] * inv;
    }
    __syncthreads();

    // Phase 3: table[b][d]
    for (int idx = t; idx < NBINS * D_; idx += 1024) {
        const int b = idx >> 9, d = idx & (D_ - 1);
        float r;
        if (b == 0) {
            r = bin_emb[d];                       // expr==0 -> bin_embeddings[0]
        } else {
            r = 0.f;
            for (int n = 0; n < 31; ++n)
                r = fmaf(probs_s[b][n], bin_emb[(size_t)(n + 1) * D_ + d], r);
        }
        table[idx] = r;
    }
}

// ---------------------------------------------------------------------------
// Kernel 3: bandwidth kernel. One wave32 per 1024-float output row.
// Row r==0: concat(cond_table[cond[c,0..3]] (4x128), batch_table[batch[c]]).
// Row r>0 : concat(gene_table[g]*(g!=0), table[expr]).
// 128-bit loads, non-temporal 128-bit stores (write-once stream, keep L2 for
// the gene_table gather). Mask written as 0/1 floats after the embedding block.
// ---------------------------------------------------------------------------
__global__ __launch_bounds__(256) void assemble_kernel(
    const int* __restrict__ gene, const int* __restrict__ expr,
    const int* __restrict__ cond, const int* __restrict__ batch,
    const int* __restrict__ pad,
    const float* __restrict__ gene_table, const float* __restrict__ cond_table,
    const float* __restrict__ batch_table, const float* __restrict__ table,
    float* __restrict__ out, float* __restrict__ mask)
{
    const int wave   = threadIdx.x >> 5;
    const int lane   = threadIdx.x & 31;
    const int rowIdx = blockIdx.x * 8 + wave;     // 0 .. 65535
    const int c      = rowIdx >> 11;
    const int r      = rowIdx & (ROWS_ - 1);
    float* orow = out + (size_t)rowIdx * 1024;
    const int doff = lane * 4;

    if (r == 0) {
#pragma unroll
        for (int k = 0; k < 4; ++k) {
            const int ci = cond[c * 4 + k];
            const f4 v = *(const f4*)(cond_table + (size_t)ci * 128 + doff);
            __builtin_nontemporal_store(v, (f4*)(orow + k * 128 + doff));
        }
        const int bi = batch[c];
        const float* brow = batch_table + (size_t)bi * D_;
#pragma unroll
        for (int k = 0; k < 4; ++k) {
            const f4 v = *(const f4*)(brow + k * 128 + doff);
            __builtin_nontemporal_store(v, (f4*)(orow + 512 + k * 128 + doff));
        }
        if (lane == 0) mask[rowIdx] = 0.f;
    } else {
        const int l = r - 1;
        const int g = gene[c * LM1_ + l];
        const int e = expr[c * LM1_ + l];
        const float gm = (g != 0) ? 1.f : 0.f;
        const float* grow = gene_table + (size_t)g * D_;
#pragma unroll
        for (int k = 0; k < 4; ++k) {
            f4 v = *(const f4*)(grow + k * 128 + doff);
            v = v * gm;
            __builtin_nontemporal_store(v, (f4*)(orow + k * 128 + doff));
        }
        const float* trow = table + (size_t)e * D_;   // L1/L2 resident (64 KB)
#pragma unroll
        for (int k = 0; k < 4; ++k) {
            const f4 v = *(const f4*)(trow + k * 128 + doff);
            __builtin_nontemporal_store(v, (f4*)(orow + 512 + k * 128 + doff));
        }
        if (lane == 0) mask[rowIdx] = (pad[c * LM1_ + l] != 0) ? 1.f : 0.f;
    }
}

// ---------------------------------------------------------------------------
extern "C" void kernel_launch(void* const* d_in, const int* in_sizes, int n_in,
                              void* d_out, int out_size, void* d_ws, size_t ws_size,
                              hipStream_t stream) {
    const int*   gene        = (const int*)d_in[0];
    const int*   expr        = (const int*)d_in[1];
    const int*   cond        = (const int*)d_in[2];
    const int*   batch       = (const int*)d_in[3];
    const int*   pad         = (const int*)d_in[4];
    const float* gene_table  = (const float*)d_in[5];
    const float* bin_emb     = (const float*)d_in[6];
    const float* W1          = (const float*)d_in[7];
    const float* b1          = (const float*)d_in[8];
    const float* W2          = (const float*)d_in[9];
    const float* b2          = (const float*)d_in[10];
    const float* cond_table  = (const float*)d_in[11];
    const float* batch_table = (const float*)d_in[12];

    float* enc   = (float*)d_ws;                    // 32*512 f32 = 64 KB
    float* table = (float*)d_ws + NBINS * D_;       // 32*512 f32 = 64 KB

    float* out  = (float*)d_out;
    float* mask = out + (size_t)C_ * ROWS_ * 1024;  // mask after embedding block

    // 1) enc = relu(b*W1 + b1) @ W2 + b2  (WMMA f32 16x16x4; 2x32 tiles)
    enc_wmma_kernel<<<64, 32, 0, stream>>>(W1, b1, W2, b2, enc);
    // 2) softmax over bins -> 32x512 expr-embedding table
    softmax_table_kernel<<<1, 1024, 0, stream>>>(enc, bin_emb, table);
    // 3) memory-bound gather/concat: one wave per output row
    assemble_kernel<<<(C_ * ROWS_) / 8, 256, 0, stream>>>(
        gene, expr, cond, batch, pad,
        gene_table, cond_table, batch_table, table, out, mask);
}